// BiLSTMClassifier_56985626083676
// MI455X (gfx1250) — compile-verified
//
#include <hip/hip_runtime.h>
#include <hip/hip_bf16.h>

// ---------------------------------------------------------------------------
// BiLSTM classifier, MI455X (gfx1250).
// Only the backward stack (b0, b1) reaches the output; forward layers are dead.
//  - input projections x@Wx+b hoisted into parallel f32-WMMA GEMMs
//    (embedding gather fused), weight panel staged in LDS via
//    GLOBAL_LOAD_ASYNC_TO_LDS_B128 (ASYNCcnt), 64 rows per workgroup.
//  - sequential scans keep h/c/Wh in LDS, batch split over 4 workgroups,
//    recurrent h@Wh via V_WMMA_F32_16X16X4_F32.
// ---------------------------------------------------------------------------

typedef __attribute__((ext_vector_type(2))) float v2f;
typedef __attribute__((ext_vector_type(8))) float v8f;

#define B_ 64
#define T_ 512
#define E_ 128
#define H_ 64
#define G_ 256 /* 4*H */
#define C_ 20

static __device__ __forceinline__ v8f wmma4(v2f a, v2f b, v8f c) {
  // D = A(16x4,f32) * B(4x16,f32) + C(16x16,f32)
  return __builtin_amdgcn_wmma_f32_16x16x4_f32(
      /*neg_a=*/false, a, /*neg_b=*/false, b,
      /*c_mod=*/(short)0, c, /*reuse_a=*/false, /*reuse_b=*/false);
}

// Async bulk copy global -> LDS (16B per lane per issue, ASYNCcnt-tracked).
// Low 32 bits of a flat LDS pointer are the raw LDS byte address.
static __device__ __forceinline__ void async_copy_b128(unsigned lds_addr,
                                                       const void* gaddr) {
  asm volatile("global_load_async_to_lds_b128 %0, %1, off" ::"v"(lds_addr),
               "v"((unsigned long long)(uintptr_t)gaddr)
               : "memory");
}
static __device__ __forceinline__ void async_wait0() {
  asm volatile("s_wait_asynccnt 0" ::: "memory");
}

// ---------------------------------------------------------------------------
// Kernel A: Z[b, s, :] = (word_emb[ids[b,t]] + pos_emb[pos[b,t]]) @ Wx + bias
// with t = T-1-s (scan order for the backward layers).
// grid = (B*T)/64 blocks, 512 threads (16 waves); wave w owns N-tile w,
// loops over 4 row tiles. Wx (128 KB) staged in LDS by async DMA.
// ---------------------------------------------------------------------------
__global__ __launch_bounds__(512) void k_embed_gemm(
    const int* __restrict__ x, const float* __restrict__ wemb,
    const float* __restrict__ pemb, const float* __restrict__ wx,
    const float* __restrict__ bias, float* __restrict__ Z) {
  extern __shared__ float smem[];
  float* WxL = smem;            // 128*256 f32 = 128 KB
  float* A = WxL + E_ * G_;     // 64*128 f32  =  32 KB
  const int row0 = blockIdx.x * 64;
  const int tid = threadIdx.x;

  // kick off weight-panel DMA
  {
    const unsigned l0 = (unsigned)(uintptr_t)WxL;
    const char* g = (const char*)wx;
    for (int off = tid * 16; off < E_ * G_ * 4; off += 512 * 16)
      async_copy_b128(l0 + off, g + off);
  }
  // embedding gather (overlaps with the DMA), float4-vectorized
  for (int i = tid; i < 64 * (E_ / 4); i += 512) {
    int r = i >> 5, k4 = (i & 31) * 4;
    int row = row0 + r;
    int b = row >> 9;        // row / T_
    int s = row & (T_ - 1);  // scan step
    int t = T_ - 1 - s;      // original time
    int id = x[b * 3 * T_ + t];
    int ps = x[b * 3 * T_ + T_ + t];
    const float4 wv = *(const float4*)(wemb + (size_t)id * E_ + k4);
    const float4 pv = *(const float4*)(pemb + (size_t)ps * E_ + k4);
    float4 o;
    o.x = wv.x + pv.x;
    o.y = wv.y + pv.y;
    o.z = wv.z + pv.z;
    o.w = wv.w + pv.w;
    *(float4*)(A + r * E_ + k4) = o;
  }
  async_wait0();
  __syncthreads();

  const int lane = tid & 31, lr = lane & 15;
  const int n0 = (tid >> 5) * 16;
  const int koff = (lane >= 16) ? 2 : 0;
  const int moff = (lane >= 16) ? 8 : 0;
  const float bv = bias[n0 + lr];
  for (int rt = 0; rt < 4; ++rt) {
    const float* Ar = A + (rt * 16 + lr) * E_;
    v8f acc = {};
    for (int k0 = 0; k0 < E_; k0 += 4) {
      v2f a, b;
      a.x = Ar[k0 + koff];
      a.y = Ar[k0 + koff + 1];
      b.x = WxL[(k0 + koff) * G_ + n0 + lr];
      b.y = WxL[(k0 + koff + 1) * G_ + n0 + lr];
      acc = wmma4(a, b, acc);
    }
#pragma unroll
    for (int r = 0; r < 8; ++r)
      Z[(size_t)(row0 + rt * 16 + moff + r) * G_ + n0 + lr] = acc[r] + bv;
  }
}

// ---------------------------------------------------------------------------
// Kernel C: Z[b, s1, :] = H0[b, T-1-s1, :] @ Wx_b1 + bias  (layer-b1 inputs;
// layer b1 consumes b0 outputs re-reversed, i.e. b0 scan index T-1-s1).
// ---------------------------------------------------------------------------
__global__ __launch_bounds__(512) void k_hidden_gemm(
    const float* __restrict__ Hin, const float* __restrict__ wx,
    const float* __restrict__ bias, float* __restrict__ Z) {
  extern __shared__ float smem[];
  float* WxL = smem;            // 64*256 f32 = 64 KB
  float* A = WxL + H_ * G_;     // 64*64 f32  = 16 KB
  const int row0 = blockIdx.x * 64;
  const int tid = threadIdx.x;

  {
    const unsigned l0 = (unsigned)(uintptr_t)WxL;
    const char* g = (const char*)wx;
    for (int off = tid * 16; off < H_ * G_ * 4; off += 512 * 16)
      async_copy_b128(l0 + off, g + off);
  }
  for (int i = tid; i < 64 * (H_ / 4); i += 512) {
    int r = i >> 4, k4 = (i & 15) * 4;
    int row = row0 + r;
    int b = row >> 9;
    int s = row & (T_ - 1);
    int src = b * T_ + (T_ - 1 - s);
    *(float4*)(A + r * H_ + k4) =
        *(const float4*)(Hin + (size_t)src * H_ + k4);
  }
  async_wait0();
  __syncthreads();

  const int lane = tid & 31, lr = lane & 15;
  const int n0 = (tid >> 5) * 16;
  const int koff = (lane >= 16) ? 2 : 0;
  const int moff = (lane >= 16) ? 8 : 0;
  const float bv = bias[n0 + lr];
  for (int rt = 0; rt < 4; ++rt) {
    const float* Ar = A + (rt * 16 + lr) * H_;
    v8f acc = {};
    for (int k0 = 0; k0 < H_; k0 += 4) {
      v2f a, b;
      a.x = Ar[k0 + koff];
      a.y = Ar[k0 + koff + 1];
      b.x = WxL[(k0 + koff) * G_ + n0 + lr];
      b.y = WxL[(k0 + koff + 1) * G_ + n0 + lr];
      acc = wmma4(a, b, acc);
    }
#pragma unroll
    for (int r = 0; r < 8; ++r)
      Z[(size_t)(row0 + rt * 16 + moff + r) * G_ + n0 + lr] = acc[r] + bv;
  }
}

// ---------------------------------------------------------------------------
// Scan kernel: one block per 16 batch rows (grid=4). h, c, Wh live in LDS.
// Per step: z = h @ Wh (WMMA, 16 waves x one 16x16 tile) + Z[s] (global),
// gate nonlinearities, masked state update, h written to Hout in scan order.
// Mask index t = T-1-s for both backward layers.
// ---------------------------------------------------------------------------
__global__ __launch_bounds__(512) void k_lstm_scan(
    const float* __restrict__ Z, const float* __restrict__ wh,
    const int* __restrict__ x, float* __restrict__ Hout) {
  extern __shared__ float smem[];
  float* WhL = smem;             // 64*256 = 64 KB
  float* hbuf = WhL + H_ * G_;   // 16*64  =  4 KB
  float* cbuf = hbuf + 16 * H_;  // 16*64  =  4 KB
  float* zbuf = cbuf + 16 * H_;  // 16*256 = 16 KB
  const int tid = threadIdx.x;
  const int b0 = blockIdx.x * 16;

  {
    const unsigned l0 = (unsigned)(uintptr_t)WhL;
    const char* g = (const char*)wh;
    for (int off = tid * 16; off < H_ * G_ * 4; off += 512 * 16)
      async_copy_b128(l0 + off, g + off);
  }
  for (int i = tid; i < 16 * H_; i += 512) {
    hbuf[i] = 0.0f;
    cbuf[i] = 0.0f;
  }
  async_wait0();
  __syncthreads();

  const int lane = tid & 31, lr = lane & 15;
  const int n0 = (tid >> 5) * 16;
  const int koff = (lane >= 16) ? 2 : 0;
  const int moff = (lane >= 16) ? 8 : 0;

  for (int s = 0; s < T_; ++s) {
    // ---- recurrent GEMM: zbuf = h @ Wh (one 16x16 tile per wave) ----
    v8f acc = {};
    for (int k0 = 0; k0 < H_; k0 += 4) {
      v2f a, b;
      a.x = hbuf[lr * H_ + k0 + koff];
      a.y = hbuf[lr * H_ + k0 + koff + 1];
      b.x = WhL[(k0 + koff) * G_ + n0 + lr];
      b.y = WhL[(k0 + koff + 1) * G_ + n0 + lr];
      acc = wmma4(a, b, acc);
    }
#pragma unroll
    for (int r = 0; r < 8; ++r) zbuf[(moff + r) * G_ + n0 + lr] = acc[r];
    // prefetch next step's precomputed input rows while we sync
    if (s + 1 < T_ && tid < 16)
      __builtin_prefetch(Z + (size_t)((b0 + tid) * T_ + s + 1) * G_, 0, 3);
    __syncthreads();

    // ---- gates + state update (1024 elems, 512 threads -> 2 each) ----
    const int t = T_ - 1 - s;
    for (int e = tid; e < 16 * H_; e += 512) {
      int br = e >> 6, j = e & (H_ - 1);
      int gb = b0 + br;
      const float* zr = Z + (size_t)(gb * T_ + s) * G_;
      float zi = zbuf[br * G_ + j] + zr[j];
      float zf = zbuf[br * G_ + H_ + j] + zr[H_ + j];
      float zg = zbuf[br * G_ + 2 * H_ + j] + zr[2 * H_ + j];
      float zo = zbuf[br * G_ + 3 * H_ + j] + zr[3 * H_ + j];
      float ig = 1.0f / (1.0f + __expf(-zi));
      float fg = 1.0f / (1.0f + __expf(-zf));
      float gg = tanhf(zg);
      float og = 1.0f / (1.0f + __expf(-zo));
      float c_old = cbuf[e], h_old = hbuf[e];
      float c_new = fg * c_old + ig * gg;
      float h_new = og * tanhf(c_new);
      int m = x[gb * 3 * T_ + 2 * T_ + t];
      float hn = m ? h_new : h_old;
      float cn = m ? c_new : c_old;
      hbuf[e] = hn;
      cbuf[e] = cn;
      Hout[(size_t)(gb * T_ + s) * H_ + j] = hn;
    }
    __syncthreads();
  }
}

// ---------------------------------------------------------------------------
// Final dense + softmax: last = H1[b, step T-1, :]; out = softmax(last@W + b)
// ---------------------------------------------------------------------------
__global__ __launch_bounds__(64) void k_dense_softmax(
    const float* __restrict__ H1, const float* __restrict__ W,
    const float* __restrict__ bias, float* __restrict__ out) {
  const int b = threadIdx.x;  // 64 threads, one batch row each
  const float* h = H1 + (size_t)(b * T_ + (T_ - 1)) * H_;
  float logits[C_];
  float mx = -3.0e38f;
#pragma unroll
  for (int c = 0; c < C_; ++c) {
    float acc = bias[c];
    for (int k = 0; k < H_; ++k) acc += h[k] * W[k * C_ + c];
    logits[c] = acc;
    mx = fmaxf(mx, acc);
  }
  float sum = 0.0f;
#pragma unroll
  for (int c = 0; c < C_; ++c) {
    logits[c] = __expf(logits[c] - mx);
    sum += logits[c];
  }
  float inv = 1.0f / sum;
#pragma unroll
  for (int c = 0; c < C_; ++c) out[b * C_ + c] = logits[c] * inv;
}

// ---------------------------------------------------------------------------
extern "C" void kernel_launch(void* const* d_in, const int* in_sizes, int n_in,
                              void* d_out, int out_size, void* d_ws,
                              size_t ws_size, hipStream_t stream) {
  const int* x = (const int*)d_in[0];
  const float* word_emb = (const float*)d_in[1];
  const float* pos_emb = (const float*)d_in[2];
  // f0/f1 weights (indices 3..8) are dead code: their outputs never reach out.
  const float* wx_b0 = (const float*)d_in[9];
  const float* wh_b0 = (const float*)d_in[10];
  const float* b_b0 = (const float*)d_in[11];
  const float* wx_b1 = (const float*)d_in[12];
  const float* wh_b1 = (const float*)d_in[13];
  const float* b_b1 = (const float*)d_in[14];
  const float* dense_w = (const float*)d_in[15];
  const float* dense_b = (const float*)d_in[16];
  float* out = (float*)d_out;

  float* Z = (float*)d_ws;               // 32768*256 f32 = 32 MB (reused)
  float* H0 = Z + (size_t)B_ * T_ * G_;  // 32768*64 f32 = 8 MB
  float* H1 = H0 + (size_t)B_ * T_ * H_; // 8 MB

  const int gemm_blocks = (B_ * T_) / 64;  // 512
  const size_t lds_embed = (size_t)(E_ * G_ + 64 * E_) * sizeof(float);
  const size_t lds_hidden = (size_t)(H_ * G_ + 64 * H_) * sizeof(float);
  const size_t lds_scan =
      (size_t)(H_ * G_ + 2 * 16 * H_ + 16 * G_) * sizeof(float);

  // layer b0
  k_embed_gemm<<<gemm_blocks, 512, lds_embed, stream>>>(x, word_emb, pos_emb,
                                                        wx_b0, b_b0, Z);
  k_lstm_scan<<<B_ / 16, 512, lds_scan, stream>>>(Z, wh_b0, x, H0);
  // layer b1 (inputs = b0 outputs re-reversed in time)
  k_hidden_gemm<<<gemm_blocks, 512, lds_hidden, stream>>>(H0, wx_b1, b_b1, Z);
  k_lstm_scan<<<B_ / 16, 512, lds_scan, stream>>>(Z, wh_b1, x, H1);
  // classifier head
  k_dense_softmax<<<1, 64, 0, stream>>>(H1, dense_w, dense_b, out);
}